// global_attention_with_topic_163208757666
// MI455X (gfx1250) — compile-verified
//
#include <hip/hip_runtime.h>
#include <hip/hip_bf16.h>

typedef __attribute__((ext_vector_type(2))) float v2f;
typedef __attribute__((ext_vector_type(8))) float v8f;

#define Bsz 32
#define Ssz 4096
#define Hsz 1024
#define Tsz 512
#define CATW 2560          // 2H + T
#define SC   32            // S-chunks per batch
#define ROWS 128           // S rows per chunk (4096/32)

__device__ __forceinline__ float wave_sum(float v) {
#pragma unroll
  for (int off = 16; off > 0; off >>= 1) v += __shfl_xor(v, off, 32);
  return v;
}
__device__ __forceinline__ float wave_max(float v) {
#pragma unroll
  for (int off = 16; off > 0; off >>= 1) v = fmaxf(v, __shfl_xor(v, off, 32));
  return v;
}

// ---------------------------------------------------------------------------
// Kernel 1: gamma = x @ W_in^T + b_in   (fp32 WMMA 16x16x4)
// One wave per 16x16 output tile. M=32 (2 tiles), N=1024 (64 tiles).
// ---------------------------------------------------------------------------
__global__ __launch_bounds__(128) void k_gamma(const float* __restrict__ x,
                                               const float* __restrict__ Win,
                                               const float* __restrict__ bin,
                                               float* __restrict__ gamma) {
  const int wave  = threadIdx.x >> 5;
  const int lane  = threadIdx.x & 31;
  const int gwave = blockIdx.x * 4 + wave;     // 0..127
  const int mt    = gwave & 1;
  const int nt    = gwave >> 1;
  const int half  = lane >> 4;                 // 0|1
  const int l16   = lane & 15;
  const int kb    = half * 2;
  const float* arow = x   + (size_t)(mt * 16 + l16) * Hsz;   // A row (per lane)
  const float* brow = Win + (size_t)(nt * 16 + l16) * Hsz;   // B=W_in^T col = W_in row
  v8f acc = {};
#pragma unroll 4
  for (int k = 0; k < Hsz; k += 4) {
    v2f a = *(const v2f*)(arow + k + kb);
    v2f b = *(const v2f*)(brow + k + kb);
    acc = __builtin_amdgcn_wmma_f32_16x16x4_f32(false, a, false, b,
                                                (short)0, acc, false, false);
  }
  const int col  = nt * 16 + l16;
  const float bv = bin[col];
#pragma unroll
  for (int r = 0; r < 8; ++r) {
    int row = mt * 16 + r + half * 8;          // C/D layout: VGPR r -> M=r / M=r+8
    gamma[(size_t)row * Hsz + col] = acc[r] + bv;
  }
}

// ---------------------------------------------------------------------------
// Kernel 2: single HBM pass over context. Per (b, chunk of 128 rows):
//   scores -> ws, chunk-local softmax partials (max, sum, acc[1024]) -> ws
// ---------------------------------------------------------------------------
__global__ __launch_bounds__(256) void k_attn_partial(const float* __restrict__ ctx,
                                                      const float* __restrict__ gamma,
                                                      float* __restrict__ scores,
                                                      float* __restrict__ pmax,
                                                      float* __restrict__ psum,
                                                      float* __restrict__ pacc) {
  const int b = blockIdx.x >> 5;
  const int c = blockIdx.x & 31;
  __shared__ float g_lds[Hsz];
  __shared__ float p_lds[ROWS];
  __shared__ float red[8];
  const int tid = threadIdx.x, wave = tid >> 5, lane = tid & 31;

  for (int i = tid; i < Hsz; i += 256) g_lds[i] = gamma[(size_t)b * Hsz + i];
  __syncthreads();

  const float* cb = ctx + ((size_t)b * Ssz + (size_t)c * ROWS) * Hsz;

  // 8 waves x 16 rows: dot(context row, gamma)
#pragma unroll 1
  for (int rr = 0; rr < 16; ++rr) {
    const int s = wave * 16 + rr;
    const float* row = cb + (size_t)s * Hsz;
    float acc = 0.f;
#pragma unroll
    for (int k = lane * 4; k < Hsz; k += 128) {
      float4 cv = *(const float4*)(row + k);
      acc += cv.x * g_lds[k] + cv.y * g_lds[k + 1] +
             cv.z * g_lds[k + 2] + cv.w * g_lds[k + 3];
    }
    acc = wave_sum(acc);
    if (lane == 0) p_lds[s] = acc;
  }
  __syncthreads();

  // chunk max + store raw scores
  float mymax = -1e30f;
  for (int s = tid; s < ROWS; s += 256) {
    float v = p_lds[s];
    scores[(size_t)b * Ssz + c * ROWS + s] = v;
    mymax = fmaxf(mymax, v);
  }
  mymax = wave_max(mymax);
  if (lane == 0) red[wave] = mymax;
  __syncthreads();
  float cmax = red[0];
#pragma unroll
  for (int i = 1; i < 8; ++i) cmax = fmaxf(cmax, red[i]);
  __syncthreads();

  // p = exp(score - cmax); chunk sum
  float mysum = 0.f;
  for (int s = tid; s < ROWS; s += 256) {
    float p = expf(p_lds[s] - cmax);
    p_lds[s] = p;
    mysum += p;
  }
  mysum = wave_sum(mysum);
  if (lane == 0) red[wave] = mysum;
  __syncthreads();
  float csum = red[0];
#pragma unroll
  for (int i = 1; i < 8; ++i) csum += red[i];

  // weighted accumulation: thread owns 4 contiguous h columns; rows come from L2
  float4 a4 = {0.f, 0.f, 0.f, 0.f};
#pragma unroll 1
  for (int s = 0; s < ROWS; ++s) {
    const float p  = p_lds[s];
    const float4 cv = *(const float4*)(cb + (size_t)s * Hsz + tid * 4);
    a4.x += p * cv.x; a4.y += p * cv.y; a4.z += p * cv.z; a4.w += p * cv.w;
  }
  if (tid == 0) { pmax[b * SC + c] = cmax; psum[b * SC + c] = csum; }
  *(float4*)(pacc + ((size_t)(b * SC + c)) * Hsz + tid * 4) = a4;
}

// ---------------------------------------------------------------------------
// Kernel 3: merge chunk partials per batch -> c_t into cat[:, 0:1024];
//           also fill cat[:,1024:2048]=x, cat[:,2048:2560]=topic; store M,L.
// ---------------------------------------------------------------------------
__global__ __launch_bounds__(256) void k_merge(const float* __restrict__ pmax,
                                               const float* __restrict__ psum,
                                               const float* __restrict__ pacc,
                                               const float* __restrict__ x,
                                               const float* __restrict__ topic,
                                               float* __restrict__ Mrow,
                                               float* __restrict__ Lrow,
                                               float* __restrict__ cat) {
  const int b = blockIdx.x, tid = threadIdx.x;
  __shared__ float sf[SC];
  float M = -1e30f;
#pragma unroll
  for (int c = 0; c < SC; ++c) M = fmaxf(M, pmax[b * SC + c]);
  float L = 0.f;
#pragma unroll
  for (int c = 0; c < SC; ++c) L += psum[b * SC + c] * expf(pmax[b * SC + c] - M);
  if (tid < SC) sf[tid] = expf(pmax[b * SC + tid] - M);
  if (tid == 0) { Mrow[b] = M; Lrow[b] = L; }
  __syncthreads();
  const float inv = 1.0f / L;
  float4 a4 = {0.f, 0.f, 0.f, 0.f};
#pragma unroll 1
  for (int c = 0; c < SC; ++c) {
    const float s = sf[c];
    const float4 pv = *(const float4*)(pacc + ((size_t)(b * SC + c)) * Hsz + tid * 4);
    a4.x += s * pv.x; a4.y += s * pv.y; a4.z += s * pv.z; a4.w += s * pv.w;
  }
  float* crow = cat + (size_t)b * CATW;
  crow[tid * 4 + 0] = a4.x * inv;
  crow[tid * 4 + 1] = a4.y * inv;
  crow[tid * 4 + 2] = a4.z * inv;
  crow[tid * 4 + 3] = a4.w * inv;
  for (int j = tid; j < Hsz; j += 256) crow[Hsz + j] = x[(size_t)b * Hsz + j];
  for (int j = tid; j < Tsz; j += 256) crow[2 * Hsz + j] = topic[(size_t)b * Tsz + j];
}

// ---------------------------------------------------------------------------
// Kernel 4: weights = exp(score - M[b]) / L[b]  -> d_out weights region
// ---------------------------------------------------------------------------
__global__ __launch_bounds__(256) void k_weights(const float* __restrict__ scores,
                                                 const float* __restrict__ Mrow,
                                                 const float* __restrict__ Lrow,
                                                 float* __restrict__ weights) {
  const int i = blockIdx.x * 256 + threadIdx.x;   // 0..131071
  const int b = i >> 12;
  weights[i] = expf(scores[i] - Mrow[b]) / Lrow[b];
}

// ---------------------------------------------------------------------------
// Kernel 5: out = tanh(cat @ W_out^T + b_out)   (fp32 WMMA, K=2560)
// ---------------------------------------------------------------------------
__global__ __launch_bounds__(128) void k_out(const float* __restrict__ cat,
                                             const float* __restrict__ Wout,
                                             const float* __restrict__ bout,
                                             float* __restrict__ out) {
  const int wave  = threadIdx.x >> 5;
  const int lane  = threadIdx.x & 31;
  const int gwave = blockIdx.x * 4 + wave;     // 0..127
  const int mt    = gwave & 1;
  const int nt    = gwave >> 1;
  const int half  = lane >> 4;
  const int l16   = lane & 15;
  const int kb    = half * 2;
  const float* arow = cat  + (size_t)(mt * 16 + l16) * CATW;
  const float* brow = Wout + (size_t)(nt * 16 + l16) * CATW;
  v8f acc = {};
#pragma unroll 4
  for (int k = 0; k < CATW; k += 4) {
    v2f a = *(const v2f*)(arow + k + kb);
    v2f b = *(const v2f*)(brow + k + kb);
    acc = __builtin_amdgcn_wmma_f32_16x16x4_f32(false, a, false, b,
                                                (short)0, acc, false, false);
  }
  const int col  = nt * 16 + l16;
  const float bv = bout[col];
#pragma unroll
  for (int r = 0; r < 8; ++r) {
    int row = mt * 16 + r + half * 8;
    out[(size_t)row * Hsz + col] = tanhf(acc[r] + bv);
  }
}

// ---------------------------------------------------------------------------
extern "C" void kernel_launch(void* const* d_in, const int* in_sizes, int n_in,
                              void* d_out, int out_size, void* d_ws, size_t ws_size,
                              hipStream_t stream) {
  const float* x     = (const float*)d_in[0];   // [32,1024]
  const float* ctx   = (const float*)d_in[1];   // [32,4096,1024]
  const float* topic = (const float*)d_in[2];   // [32,512]
  const float* Win   = (const float*)d_in[3];   // [1024,1024]
  const float* bin   = (const float*)d_in[4];   // [1024]
  const float* Wout  = (const float*)d_in[5];   // [1024,2560]
  const float* bout  = (const float*)d_in[6];   // [1024]

  float* out     = (float*)d_out;               // [32,1024]
  float* weights = out + (size_t)Bsz * Hsz;     // [32,4096]

  // workspace layout (floats)
  float* ws     = (float*)d_ws;
  float* gamma  = ws;                                   // 32*1024
  float* scores = gamma  + (size_t)Bsz * Hsz;           // 32*4096
  float* pmax   = scores + (size_t)Bsz * Ssz;           // 32*32
  float* psum   = pmax   + Bsz * SC;                    // 32*32
  float* Mrow   = psum   + Bsz * SC;                    // 32
  float* Lrow   = Mrow   + Bsz;                         // 32
  float* cat    = Lrow   + Bsz;                         // 32*2560
  float* pacc   = cat    + (size_t)Bsz * CATW;          // 32*32*1024 (4 MB)

  k_gamma<<<32, 128, 0, stream>>>(x, Win, bin, gamma);
  k_attn_partial<<<Bsz * SC, 256, 0, stream>>>(ctx, gamma, scores, pmax, psum, pacc);
  k_merge<<<Bsz, 256, 0, stream>>>(pmax, psum, pacc, x, topic, Mrow, Lrow, cat);
  k_weights<<<(Bsz * Ssz) / 256, 256, 0, stream>>>(scores, Mrow, Lrow, weights);
  k_out<<<32, 128, 0, stream>>>(cat, Wout, bout, out);
}